// GCN_50672024159115
// MI455X (gfx1250) — compile-verified
//
#include <hip/hip_runtime.h>
#include <hip/hip_bf16.h>

#define NUSER 50000
#define NITEM 50000
#define NNODE (NUSER + NITEM)
#define NEDGE 1600000
#define DFEAT 128
#define DIM   64
#define L2EPS 1e-12f

typedef __attribute__((ext_vector_type(2))) float v2f;
typedef __attribute__((ext_vector_type(8))) float v8f;

__device__ __forceinline__ float lrelu(float v) { return v > 0.f ? v : 0.01f * v; }

// ---------------------------------------------------------------------------
// WMMA GEMM: out[M x 64] = A[M x K] @ W[K x 64] (+bias)(+lrelu)(accumulate)
// One wave computes a 16-row x 64-col tile with 4 f32 16x16 accumulators.
// W is staged in LDS once per block (16-32 KB of the 320 KB/WGP pool).
// MODE: 0 = plain, 1 = +bias, 2 = +bias then lrelu, 3 = out += (A@W + bias)
// ---------------------------------------------------------------------------
template <int K, int MODE>
__global__ void gemm64_wmma_kernel(const float* __restrict__ A,
                                   const float* __restrict__ W,
                                   const float* __restrict__ bias,
                                   float* __restrict__ out, int M) {
    __shared__ float wlds[K * DIM];

    // cooperative stage of W into LDS (float4 per thread per step)
    for (int i = threadIdx.x; i < (K * DIM) / 4; i += blockDim.x)
        ((float4*)wlds)[i] = ((const float4*)W)[i];
    __syncthreads();

    int wave = (int)((blockIdx.x * blockDim.x + threadIdx.x) >> 5);
    int lane = threadIdx.x & 31;
    int nTiles = M >> 4;
    if (wave >= nTiles) return;          // wave-uniform: EXEC stays all-ones
    int rowBase = wave << 4;
    int hf = lane >> 4;                  // half-wave: 0 -> K+0/K+1, 1 -> K+2/K+3
    int lm = lane & 15;

    const float* arow = A + (size_t)(rowBase + lm) * K + 2 * hf;
    v8f acc0 = {}, acc1 = {}, acc2 = {}, acc3 = {};

#pragma unroll
    for (int k0 = 0; k0 < K; k0 += 4) {
        v2f a;
        a.x = arow[k0];                  // A[row][k0 + 2*hf]
        a.y = arow[k0 + 1];              // A[row][k0 + 2*hf + 1]
        const float* wrow = wlds + (k0 + 2 * hf) * DIM + lm;
        v2f b0, b1, b2, b3;
        b0.x = wrow[0];       b0.y = wrow[DIM];
        b1.x = wrow[16];      b1.y = wrow[DIM + 16];
        b2.x = wrow[32];      b2.y = wrow[DIM + 32];
        b3.x = wrow[48];      b3.y = wrow[DIM + 48];
        acc0 = __builtin_amdgcn_wmma_f32_16x16x4_f32(false, a, false, b0, (short)0, acc0, false, false);
        acc1 = __builtin_amdgcn_wmma_f32_16x16x4_f32(false, a, false, b1, (short)0, acc1, false, false);
        acc2 = __builtin_amdgcn_wmma_f32_16x16x4_f32(false, a, false, b2, (short)0, acc2, false, false);
        acc3 = __builtin_amdgcn_wmma_f32_16x16x4_f32(false, a, false, b3, (short)0, acc3, false, false);
    }

#pragma unroll
    for (int nt = 0; nt < 4; ++nt) {
        v8f acc = (nt == 0) ? acc0 : (nt == 1) ? acc1 : (nt == 2) ? acc2 : acc3;
        int col = nt * 16 + lm;
        float bv = (MODE != 0) ? bias[col] : 0.f;
#pragma unroll
        for (int v = 0; v < 8; ++v) {
            int m = v + 8 * hf;          // C/D layout: VGPR v -> M=v (+8 upper half)
            float val = acc[v] + bv;
            if (MODE == 2) val = lrelu(val);
            size_t idx = (size_t)(rowBase + m) * DIM + col;
            if (MODE == 3) out[idx] += val;
            else           out[idx] = val;
        }
    }
}

// ---------------------------------------------------------------------------
// Degree / dinv
// ---------------------------------------------------------------------------
__global__ void fill_kernel(float* __restrict__ p, float v, int n) {
    int i = blockIdx.x * blockDim.x + threadIdx.x;
    if (i < n) p[i] = v;
}

__global__ void deg_accum_kernel(const long long* __restrict__ src,
                                 float* __restrict__ deg, int e) {
    int i = blockIdx.x * blockDim.x + threadIdx.x;
    if (i < e) atomicAdd(&deg[(int)src[i]], 1.0f);
}

__global__ void rsqrt_kernel(float* __restrict__ d, int n) {
    int i = blockIdx.x * blockDim.x + threadIdx.x;
    if (i < n) d[i] = rsqrtf(d[i]);      // deg >= 1 (self-loops)
}

// ---------------------------------------------------------------------------
// One-time edge preprocess: int64 -> int32 indices + fused norm coefficient.
// Cuts per-conv edge metadata traffic in half (reused by all 4 convs).
// ---------------------------------------------------------------------------
__global__ void edge_prep_kernel(const long long* __restrict__ src64,
                                 const long long* __restrict__ dst64,
                                 const float* __restrict__ dinv,
                                 int* __restrict__ esrc, int* __restrict__ edst,
                                 float* __restrict__ ecoef, int e) {
    int i = blockIdx.x * blockDim.x + threadIdx.x;
    if (i >= e) return;
    int s = (int)src64[i];
    int d = (int)dst64[i];
    esrc[i] = s;
    edst[i] = d;
    ecoef[i] = dinv[s] * dinv[d];
}

// ---------------------------------------------------------------------------
// agg[i][:] = dinv[i]^2 * xw[i][:]   (self-loop term initializes the buffer)
// ---------------------------------------------------------------------------
__global__ void selfloop_init_kernel(const float* __restrict__ xw,
                                     const float* __restrict__ dinv,
                                     float* __restrict__ agg, int nrows) {
    int i = blockIdx.x * blockDim.x + threadIdx.x;
    int row = i >> 6;
    if (row >= nrows) return;
    float d = dinv[row];
    agg[i] = d * d * xw[i];
}

// ---------------------------------------------------------------------------
// Edge scatter: agg[dst] += coef * xw[src]; 16 lanes per edge,
// float4 gather per lane -> 4 f32 atomics (resolved in L2).
// ---------------------------------------------------------------------------
__global__ void edge_scatter_kernel(const int* __restrict__ esrc,
                                    const int* __restrict__ edst,
                                    const float* __restrict__ ecoef,
                                    const float* __restrict__ xw,
                                    float* __restrict__ agg, int e) {
    int gid = blockIdx.x * blockDim.x + threadIdx.x;
    int edge = gid >> 4;
    if (edge >= e) return;
    int part = gid & 15;
    int s = esrc[edge];
    int d = edst[edge];
    float c = ecoef[edge];
    float4 v = ((const float4*)(xw + (size_t)s * DIM))[part];
    float* o = agg + (size_t)d * DIM + part * 4;
    atomicAdd(o + 0, c * v.x);
    atomicAdd(o + 1, c * v.y);
    atomicAdd(o + 2, c * v.z);
    atomicAdd(o + 3, c * v.w);
}

// ---------------------------------------------------------------------------
// Row-wise L2 normalize in place: one wave per 64-float row (wave32).
// ---------------------------------------------------------------------------
__global__ void l2norm_rows_kernel(float* __restrict__ x, int nrows) {
    int wave = (int)((blockIdx.x * blockDim.x + threadIdx.x) >> 5);
    int lane = threadIdx.x & 31;
    if (wave >= nrows) return;
    float2* row = (float2*)(x + (size_t)wave * DIM);
    float2 v = row[lane];
    float s = v.x * v.x + v.y * v.y;
#pragma unroll
    for (int m = 16; m >= 1; m >>= 1) s += __shfl_xor(s, m, 32);
    float inv = 1.0f / fmaxf(sqrtf(s), L2EPS);
    v.x *= inv; v.y *= inv;
    row[lane] = v;
}

// ---------------------------------------------------------------------------
// out = lrelu(l2norm(agg + bias)) : fused conv epilogue, one wave per row.
// ---------------------------------------------------------------------------
__global__ void post_conv_kernel(const float* __restrict__ agg,
                                 const float* __restrict__ bias,
                                 float* __restrict__ out, int nrows) {
    int wave = (int)((blockIdx.x * blockDim.x + threadIdx.x) >> 5);
    int lane = threadIdx.x & 31;
    if (wave >= nrows) return;
    float2 v = ((const float2*)(agg + (size_t)wave * DIM))[lane];
    float2 b = ((const float2*)bias)[lane];
    v.x += b.x; v.y += b.y;
    float s = v.x * v.x + v.y * v.y;
#pragma unroll
    for (int m = 16; m >= 1; m >>= 1) s += __shfl_xor(s, m, 32);
    float inv = 1.0f / fmaxf(sqrtf(s), L2EPS);
    v.x = lrelu(v.x * inv);
    v.y = lrelu(v.y * inv);
    ((float2*)(out + (size_t)wave * DIM))[lane] = v;
}

// ---------------------------------------------------------------------------
// Host orchestration
// ---------------------------------------------------------------------------
static inline int gemm_blocks(int M) { int waves = M >> 4; return (waves * 32 + 255) / 256; }
static inline int row_wave_blocks(int rows) { return (rows * 32 + 255) / 256; }

extern "C" void kernel_launch(void* const* d_in, const int* in_sizes, int n_in,
                              void* d_out, int out_size, void* d_ws, size_t ws_size,
                              hipStream_t stream) {
    const float* features  = (const float*)d_in[0];
    const float* pref      = (const float*)d_in[1];
    const long long* eidx  = (const long long*)d_in[2];
    const long long* src64 = eidx;
    const long long* dst64 = eidx + NEDGE;
    const float* mlp_w  = (const float*)d_in[3];
    const float* mlp_b  = (const float*)d_in[4];
    const float* conv1_w = (const float*)d_in[5],  *conv1_b = (const float*)d_in[6];
    const float* g1_w    = (const float*)d_in[7],  *g1_b    = (const float*)d_in[8];
    const float* conv2_w = (const float*)d_in[9],  *conv2_b = (const float*)d_in[10];
    const float* g2_w    = (const float*)d_in[11], *g2_b    = (const float*)d_in[12];
    const float* conv4_w = (const float*)d_in[13], *conv4_b = (const float*)d_in[14];
    const float* lin4_w  = (const float*)d_in[15], *lin4_b  = (const float*)d_in[16];
    const float* g4_w    = (const float*)d_in[17], *g4_b    = (const float*)d_in[18];
    const float* conv5_w = (const float*)d_in[19], *conv5_b = (const float*)d_in[20];
    const float* lin5_w  = (const float*)d_in[21], *lin5_b  = (const float*)d_in[22];
    const float* g5_w    = (const float*)d_in[23], *g5_b    = (const float*)d_in[24];

    // workspace: x[N*64] | h[N*64] | agg[N*64] | dinv[N] | esrc[E] | edst[E] | ecoef[E]
    float* x    = (float*)d_ws;
    float* h    = x + (size_t)NNODE * DIM;
    float* agg  = h + (size_t)NNODE * DIM;
    float* dinv = agg + (size_t)NNODE * DIM;
    int*   esrc = (int*)(dinv + NNODE);
    int*   edst = esrc + NEDGE;
    float* ecoef = (float*)(edst + NEDGE);

    float* mu = (float*)d_out;
    float* lv = mu + (size_t)NNODE * DIM;

    const int T = 256;

    // ---- degree -> dinv, then fused per-edge coefficients (shared by 4 convs) ----
    fill_kernel<<<(NNODE + T - 1) / T, T, 0, stream>>>(dinv, 1.0f, NNODE);
    deg_accum_kernel<<<(NEDGE + T - 1) / T, T, 0, stream>>>(src64, dinv, NEDGE);
    rsqrt_kernel<<<(NNODE + T - 1) / T, T, 0, stream>>>(dinv, NNODE);
    edge_prep_kernel<<<(NEDGE + T - 1) / T, T, 0, stream>>>(
        src64, dst64, dinv, esrc, edst, ecoef, NEDGE);

    // ---- x = l2norm(concat(preference, features @ mlp_w + mlp_b)) ----
    hipMemcpyAsync(x, pref, (size_t)NUSER * DIM * sizeof(float),
                   hipMemcpyDeviceToDevice, stream);
    gemm64_wmma_kernel<DFEAT, 1><<<gemm_blocks(NITEM), T, 0, stream>>>(
        features, mlp_w, mlp_b, x + (size_t)NUSER * DIM, NITEM);
    l2norm_rows_kernel<<<row_wave_blocks(NNODE), T, 0, stream>>>(x, NNODE);

    const int gB  = gemm_blocks(NNODE);
    const int rB  = row_wave_blocks(NNODE);
    const int eB  = (NEDGE * 16 + T - 1) / T;
    const int sB  = (NNODE * DIM + T - 1) / T;

    // conv pipeline: h <- lrelu(l2norm(scatter(dinv-scaled x@W) + b))
    auto conv = [&](const float* W, const float* b) {
        gemm64_wmma_kernel<DIM, 0><<<gB, T, 0, stream>>>(x, W, nullptr, h, NNODE);
        selfloop_init_kernel<<<sB, T, 0, stream>>>(h, dinv, agg, NNODE);
        edge_scatter_kernel<<<eB, T, 0, stream>>>(esrc, edst, ecoef, h, agg, NEDGE);
        post_conv_kernel<<<rB, T, 0, stream>>>(agg, b, h, NNODE);
    };

    // ---- layer 1 ----
    conv(conv1_w, conv1_b);
    gemm64_wmma_kernel<DIM, 2><<<gB, T, 0, stream>>>(h, g1_w, g1_b, x, NNODE);
    // ---- layer 2 ----
    conv(conv2_w, conv2_b);
    gemm64_wmma_kernel<DIM, 2><<<gB, T, 0, stream>>>(h, g2_w, g2_b, x, NNODE);

    // ---- mu head: mu = (conv_h @ g4 + g4_b) + lrelu(x @ lin4 + lin4_b) ----
    conv(conv4_w, conv4_b);
    gemm64_wmma_kernel<DIM, 2><<<gB, T, 0, stream>>>(x, lin4_w, lin4_b, mu, NNODE);
    gemm64_wmma_kernel<DIM, 3><<<gB, T, 0, stream>>>(h, g4_w, g4_b, mu, NNODE);

    // ---- logvar head ----
    conv(conv5_w, conv5_b);
    gemm64_wmma_kernel<DIM, 2><<<gB, T, 0, stream>>>(x, lin5_w, lin5_b, lv, NNODE);
    gemm64_wmma_kernel<DIM, 3><<<gB, T, 0, stream>>>(h, g5_w, g5_b, lv, NNODE);
}